// MMD_loss_74560632259140
// MI455X (gfx1250) — compile-verified
//
#include <hip/hip_runtime.h>

typedef __attribute__((ext_vector_type(2))) float v2f;
typedef __attribute__((ext_vector_type(8))) float v8f;

#define N_ROWS 4096
#define D_DIM  1024
#define M_TOT  8192          // 2*N_ROWS
#define MMD_KERNEL_MUL 2
#define MMD_KERNEL_NUM 5
#define MMD_EPS 1e-6f

// ws layout (floats): [0 .. M_TOT)   = sq[i] = sum_d total[i,d]^2
//                     [M_TOT]        = sum(L2)           (pass 0)
//                     [M_TOT+1]      = weighted kernel sum (pass 1)

__device__ __forceinline__ const float* row_ptr(const float* z, const float* p, int row) {
    return (row < N_ROWS) ? (z + (size_t)row * D_DIM)
                          : (p + (size_t)(row - N_ROWS) * D_DIM);
}

// One wave per row: sq[row] = ||total[row]||^2. Also zeroes the two accumulators.
__global__ void mmd_sq_kernel(const float* __restrict__ z,
                              const float* __restrict__ p,
                              float* __restrict__ ws) {
    const int lane = threadIdx.x & 31;
    const int wave = threadIdx.x >> 5;
    const int row  = blockIdx.x * 8 + wave;
    if (blockIdx.x == 0 && threadIdx.x == 0) { ws[M_TOT] = 0.f; ws[M_TOT + 1] = 0.f; }
    if (row >= M_TOT) return;
    const float* r = row_ptr(z, p, row);
    float s = 0.f;
    #pragma unroll 8
    for (int k = lane; k < D_DIM; k += 32) { float v = r[k]; s += v * v; }
    #pragma unroll
    for (int off = 16; off > 0; off >>= 1) s += __shfl_xor(s, off);
    if (lane == 0) ws[row] = s;
}

// Fused GEMM pass. Each wave computes a 16x64 strip of gram = total @ total^T
// with V_WMMA_F32_16X16X4_F32, then:
//   PASS==0: accumulates sum(L2) into ws[M_TOT]
//   PASS==1: accumulates block-weighted sum of the 5-kernel exp sums into ws[M_TOT+1]
template <int PASS>
__global__ void mmd_pass_kernel(const float* __restrict__ z,
                                const float* __restrict__ p,
                                float* __restrict__ ws) {
    const int lane   = threadIdx.x & 31;
    const int waveId = (int)((blockIdx.x * blockDim.x + threadIdx.x) >> 5);

    const int J_GROUPS = M_TOT / 64;              // 128 groups of 4 N-tiles
    const int itile = waveId / J_GROUPS;          // 0..511
    const int i0    = itile * 16;
    const int j0    = (waveId % J_GROUPS) * 64;

    // f32 16x16x4 A/B fragment: lane holds row (lane&15), K pair at (lane>>4)*2.
    // For T*T^T both operands are rows of `total`, identical per-lane layout.
    const int mrow = lane & 15;
    const int koff = (lane >> 4) * 2;

    const float* aP  = row_ptr(z, p, i0 + mrow)      + koff;
    const float* bP0 = row_ptr(z, p, j0 +  0 + mrow) + koff;
    const float* bP1 = row_ptr(z, p, j0 + 16 + mrow) + koff;
    const float* bP2 = row_ptr(z, p, j0 + 32 + mrow) + koff;
    const float* bP3 = row_ptr(z, p, j0 + 48 + mrow) + koff;

    v8f c0 = {}, c1 = {}, c2 = {}, c3 = {};
    #pragma unroll 2
    for (int k = 0; k < D_DIM; k += 4) {
        v2f a  = *(const v2f*)(aP  + k);
        v2f b0 = *(const v2f*)(bP0 + k);
        v2f b1 = *(const v2f*)(bP1 + k);
        v2f b2 = *(const v2f*)(bP2 + k);
        v2f b3 = *(const v2f*)(bP3 + k);
        c0 = __builtin_amdgcn_wmma_f32_16x16x4_f32(false, a, false, b0, (short)0, c0, false, false);
        c1 = __builtin_amdgcn_wmma_f32_16x16x4_f32(false, a, false, b1, (short)0, c1, false, false);
        c2 = __builtin_amdgcn_wmma_f32_16x16x4_f32(false, a, false, b2, (short)0, c2, false, false);
        c3 = __builtin_amdgcn_wmma_f32_16x16x4_f32(false, a, false, b3, (short)0, c3, false, false);
    }

    // C/D layout: VGPR r, lanes 0-15 -> (M=r, N=lane); lanes 16-31 -> (M=r+8, N=lane-16)
    const float* sq   = ws;
    const int   ibase = i0 + ((lane >> 4) << 3);
    const int   jlane = lane & 15;
    const bool  izSide = (i0 < N_ROWS);

    float sqj[4], wt[4];
    #pragma unroll
    for (int t = 0; t < 4; ++t) {
        sqj[t] = sq[j0 + 16 * t + jlane];
        wt[t]  = (((j0 + 16 * t) < N_ROWS) == izSide) ? 1.f : -1.f;
    }

    float invbw[MMD_KERNEL_NUM];
    if constexpr (PASS == 1) {
        const float sumL2 = ws[M_TOT];
        const float denom = (float)((double)M_TOT * (double)M_TOT - (double)M_TOT);
        float bw = fmaxf(sumL2 / denom, MMD_EPS) * 0.25f;  // / KERNEL_MUL^(KERNEL_NUM/2)
        #pragma unroll
        for (int q = 0; q < MMD_KERNEL_NUM; ++q) {
            invbw[q] = 1.0f / fmaxf(bw, MMD_EPS);
            bw *= (float)MMD_KERNEL_MUL;
        }
    }

    float lsum = 0.f;
    #pragma unroll
    for (int r = 0; r < 8; ++r) {
        const float si = sq[ibase + r];
        const float l0 = fmaxf(si + sqj[0] - 2.f * c0[r], 0.f);
        const float l1 = fmaxf(si + sqj[1] - 2.f * c1[r], 0.f);
        const float l2 = fmaxf(si + sqj[2] - 2.f * c2[r], 0.f);
        const float l3 = fmaxf(si + sqj[3] - 2.f * c3[r], 0.f);
        if constexpr (PASS == 0) {
            lsum += (l0 + l1) + (l2 + l3);
        } else {
            float k0 = 0.f, k1 = 0.f, k2 = 0.f, k3 = 0.f;
            #pragma unroll
            for (int q = 0; q < MMD_KERNEL_NUM; ++q) {
                k0 += __expf(-l0 * invbw[q]);
                k1 += __expf(-l1 * invbw[q]);
                k2 += __expf(-l2 * invbw[q]);
                k3 += __expf(-l3 * invbw[q]);
            }
            lsum += wt[0] * k0 + wt[1] * k1 + wt[2] * k2 + wt[3] * k3;
        }
    }

    #pragma unroll
    for (int off = 16; off > 0; off >>= 1) lsum += __shfl_xor(lsum, off);
    if (lane == 0) atomicAdd(&ws[M_TOT + PASS], lsum);
}

__global__ void mmd_finalize_kernel(const float* __restrict__ ws, float* __restrict__ out) {
    const float mmd = ws[M_TOT + 1] / ((float)N_ROWS * (float)N_ROWS);
    out[0] = fmaxf(mmd, 0.f);
}

extern "C" void kernel_launch(void* const* d_in, const int* in_sizes, int n_in,
                              void* d_out, int out_size, void* d_ws, size_t ws_size,
                              hipStream_t stream) {
    (void)in_sizes; (void)n_in; (void)out_size; (void)ws_size;
    const float* z = (const float*)d_in[0];
    const float* p = (const float*)d_in[1];
    float* out = (float*)d_out;
    float* ws  = (float*)d_ws;

    // 1) row squared-norms + zero accumulators
    mmd_sq_kernel<<<M_TOT / 8, 256, 0, stream>>>(z, p, ws);

    // 2) two fused WMMA GEMM passes over the 8192x8192 gram matrix
    //    512 itiles * 128 jgroups = 65536 waves; 8 waves per 256-thread block
    const int blocks = (512 * 128) / 8;   // 8192
    mmd_pass_kernel<0><<<blocks, 256, 0, stream>>>(z, p, ws);
    mmd_pass_kernel<1><<<blocks, 256, 0, stream>>>(z, p, ws);

    // 3) scalar finalize
    mmd_finalize_kernel<<<1, 1, 0, stream>>>(ws, out);
}